// Quantize_57354993271279
// MI455X (gfx1250) — compile-verified
//
#include <hip/hip_runtime.h>
#include <math.h>

// Problem constants
#define BATCH 32
#define TLEN 512
#define IDIM 512          // input dim / FFT length
#define VQ 128            // feature dim
#define CBNUM 8           // codebooks
#define NEMB 512          // entries per codebook
#define ROWS (BATCH * TLEN)   // 16384

typedef float v2f __attribute__((ext_vector_type(2)));
typedef float v8f __attribute__((ext_vector_type(8)));

__device__ __forceinline__ v8f wmma_f32_4(v2f a, v2f b, v8f c) {
  // D = A(16x4 f32) * B(4x16 f32) + C(16x16 f32)
  return __builtin_amdgcn_wmma_f32_16x16x4_f32(
      false, a, false, b, (short)0, c, false, false);
}

__device__ __forceinline__ int bitrev9(int n) {
  return (int)(__brev((unsigned)n) >> 23);
}

// In-place radix-2 DIT FFT of length 512 on LDS arrays already in
// bit-reversed order. 256 threads, one butterfly each per stage.
// sign = -1: forward DFT, sign = +1: inverse DFT kernel (caller scales).
__device__ __forceinline__ void fft512(float* re, float* im, int tid, float sign) {
#pragma unroll
  for (int s = 1; s <= 9; ++s) {
    const int m = 1 << s;
    const int half = m >> 1;
    const int j = tid & (half - 1);
    const int base = (tid >> (s - 1)) * m;
    const float ang = sign * 6.28318530717958647692f * (float)j / (float)m;
    const float wr = cosf(ang);
    const float wi = sinf(ang);
    const int i1 = base + j;
    const int i2 = i1 + half;
    const float br = re[i2], bi = im[i2];
    const float tr = wr * br - wi * bi;
    const float ti = wr * bi + wi * br;
    const float ur = re[i1], ui = im[i1];
    re[i2] = ur - tr; im[i2] = ui - ti;
    re[i1] = ur + tr; im[i1] = ui + ti;
    __syncthreads();
  }
}

// ---------------------------------------------------------------------------
// Kernel 1: per-row rfft -> phase shift by -k*angle(X1) -> irfft
// One block (256 threads) per row. Output x_recon row (512 f32) to ws.
// ---------------------------------------------------------------------------
__global__ void __launch_bounds__(256) fft_rotate_kernel(
    const float* __restrict__ x, float* __restrict__ xrec) {
  __shared__ float re[512], im[512];
  __shared__ float zre[512], zim[512];
  const int tid = threadIdx.x;
  const int row = blockIdx.x;
  const float* xr = x + (size_t)row * IDIM;

  // Load input at bit-reversed positions (imag = 0)
#pragma unroll
  for (int i = 0; i < 2; ++i) {
    const int n = tid * 2 + i;
    const int b = bitrev9(n);
    re[b] = xr[n];
    im[b] = 0.0f;
  }
  __syncthreads();

  fft512(re, im, tid, -1.0f);   // forward; ends with syncthreads

  // Rotation angle from bin 1 (same for all threads)
  const float p1 = atan2f(im[1], re[1]);

  // Y[k] = X[k] * e^{-i k p1}, k = 0..256; drop Im at k=0,256 (irfft
  // semantics); scatter Hermitian mirror, all at bit-reversed positions.
  for (int k = tid; k < 257; k += 256) {
    const float xre = re[k], xim = im[k];
    const float ang = -p1 * (float)k;
    const float c = cosf(ang), s = sinf(ang);
    float yr = xre * c - xim * s;
    float yi = xre * s + xim * c;
    if (k == 0 || k == 256) yi = 0.0f;
    const int bd = bitrev9(k);
    zre[bd] = yr; zim[bd] = yi;
    if (k >= 1 && k <= 255) {
      const int bm = bitrev9(512 - k);
      zre[bm] = yr; zim[bm] = -yi;
    }
  }
  __syncthreads();

  fft512(zre, zim, tid, +1.0f); // inverse (unscaled)

  float* out = xrec + (size_t)row * IDIM;
#pragma unroll
  for (int i = 0; i < 2; ++i) {
    const int n = tid + 256 * i;
    out[n] = zre[n] * (1.0f / 512.0f);
  }
}

// ---------------------------------------------------------------------------
// Kernel 2: normalize codebook rows. One wave per row (4096 rows).
// ---------------------------------------------------------------------------
__global__ void __launch_bounds__(256) cbnorm_kernel(
    const float* __restrict__ cb, float* __restrict__ cbN) {
  const int wave = threadIdx.x >> 5;
  const int lane = threadIdx.x & 31;
  const int row = blockIdx.x * 8 + wave;   // < 4096
  const float* src = cb + (size_t)row * VQ;
  float v[4];
  float ss = 0.0f;
#pragma unroll
  for (int j = 0; j < 4; ++j) {
    v[j] = src[lane * 4 + j];
    ss += v[j] * v[j];
  }
#pragma unroll
  for (int m = 16; m >= 1; m >>= 1) ss += __shfl_xor(ss, m, 32);
  const float rn = 1.0f / sqrtf(ss);
  float* dst = cbN + (size_t)row * VQ;
#pragma unroll
  for (int j = 0; j < 4; ++j) dst[lane * 4 + j] = v[j] * rn;
}

// ---------------------------------------------------------------------------
// Kernel 3: feature = x_recon @ projector  (M=16384, K=512, N=128), fp32 WMMA.
// Block = 64 rows x 128 cols; K staged in 4 chunks of 128 through LDS.
// 8 waves = 4 row-tiles (16 rows) x 2 col-halves (64 cols = 4 N-tiles).
// One A fragment is shared across the wave's 4 WMMAs per k-step.
// (Feature normalization is skipped: it cannot change the argmax.)
// ---------------------------------------------------------------------------
#define KCH 128        // K chunk staged in LDS
#define APITCH 132     // 128 + 4 pad floats -> conflict-free column reads
__global__ void __launch_bounds__(256) proj_gemm_kernel(
    const float* __restrict__ xrec, const float* __restrict__ proj,
    float* __restrict__ feat) {
  __shared__ float As[64 * APITCH];
  const int tid = threadIdx.x;
  const int wave = tid >> 5;
  const int lane = tid & 31;
  const int rowbase = blockIdx.x * 64;
  const int rowtile = wave >> 1;        // 0..3 : 16-row tile
  const int nhalf = (wave & 1) * 64;    // 0 or 64 : 4 N-tiles of 16

  const int l15 = lane & 15;
  const int khi = (lane >> 4) * 2;      // K offset 0 or 2 per half-wave

  v8f acc[4] = {};
  for (int kc = 0; kc < 512; kc += KCH) {
    // Stage 64 rows x 128 K-chunk of A into LDS (coalesced)
    __syncthreads();
    for (int idx = tid; idx < 64 * KCH; idx += 256) {
      const int r = idx >> 7, d = idx & 127;
      As[r * APITCH + d] = xrec[(size_t)(rowbase + r) * IDIM + kc + d];
    }
    __syncthreads();

    const float* arow = As + (rowtile * 16 + l15) * APITCH;
    for (int kb = 0; kb < KCH; kb += 4) {
      v2f a;
#pragma unroll
      for (int j = 0; j < 2; ++j) a[j] = arow[kb + j + khi];
#pragma unroll
      for (int t = 0; t < 4; ++t) {
        v2f b;
#pragma unroll
        for (int j = 0; j < 2; ++j) {
          const int k = kc + kb + j + khi;
          b[j] = proj[(size_t)k * VQ + nhalf + t * 16 + l15];
        }
        acc[t] = wmma_f32_4(a, b, acc[t]);
      }
    }
  }

  const int rr = rowbase + rowtile * 16 + (lane >> 4) * 8;  // rows 0-7 / 8-15
#pragma unroll
  for (int t = 0; t < 4; ++t)
#pragma unroll
    for (int r = 0; r < 8; ++r)
      feat[(size_t)(rr + r) * VQ + nhalf + t * 16 + l15] = acc[t][r];
}

// ---------------------------------------------------------------------------
// Kernel 4: similarity + argmax. Block = 64 rows x one codebook c.
// sim[row, m] = sum_d feat[row,d] * cbN[c,m,d]; argmax over m (512).
// 8 waves = 4 row-tiles x 2 N-halves; each wave sweeps 256 entries in
// 16 passes of 16-wide fp32 WMMA tiles, keeping per-lane running argmax.
// ---------------------------------------------------------------------------
#define FPITCH 132   // 128 + 4 pad
__global__ void __launch_bounds__(256) sim_argmax_kernel(
    const float* __restrict__ feat, const float* __restrict__ cbN,
    int* __restrict__ out) {
  __shared__ float Fs[64 * FPITCH];
  __shared__ float wv[2][64];
  __shared__ int   wi[2][64];
  const int tid = threadIdx.x;
  const int wave = tid >> 5;
  const int lane = tid & 31;
  const int rowbase = blockIdx.x * 64;
  const int c = blockIdx.y;
  const int rowtile = wave >> 1;        // 0..3
  const int nhalf = wave & 1;           // 0: entries 0-255, 1: 256-511

  for (int idx = tid; idx < 64 * VQ; idx += 256) {
    const int r = idx >> 7, d = idx & 127;
    Fs[r * FPITCH + d] = feat[(size_t)(rowbase + r) * VQ + d];
  }
  __syncthreads();

  const int l15 = lane & 15;
  const int khi = (lane >> 4) * 2;
  const float* cbc = cbN + (size_t)c * NEMB * VQ;
  const float* arow = Fs + (rowtile * 16 + l15) * FPITCH;

  float best[8];
  int bidx[8];
#pragma unroll
  for (int r = 0; r < 8; ++r) { best[r] = -INFINITY; bidx[r] = 0; }

  for (int pass = 0; pass < 16; ++pass) {
    const int nb = nhalf * 256 + pass * 16;    // codebook entry base
    const int n = nb + l15;                    // this lane's column (m)
    v8f acc = {};
    for (int kb = 0; kb < VQ; kb += 4) {
      v2f a, b;
#pragma unroll
      for (int j = 0; j < 2; ++j) {
        const int k = kb + j + khi;
        a[j] = arow[k];
        b[j] = cbc[(size_t)n * VQ + k];
      }
      acc = wmma_f32_4(a, b, acc);
    }
    // Per-lane running argmax (idx strictly increases across passes, so
    // strict > keeps the first occurrence on ties).
#pragma unroll
    for (int r = 0; r < 8; ++r) {
      const float v = acc[r];
      if (v > best[r]) { best[r] = v; bidx[r] = n; }
    }
  }

  // Reduce across the 16 lanes that hold the same row set
  // (lanes 0-15: rows 0-7; lanes 16-31: rows 8-15).
#pragma unroll
  for (int r = 0; r < 8; ++r) {
#pragma unroll
    for (int m = 1; m <= 8; m <<= 1) {
      const float ov = __shfl_xor(best[r], m, 32);
      const int   oi = __shfl_xor(bidx[r], m, 32);
      if (ov > best[r] || (ov == best[r] && oi < bidx[r])) {
        best[r] = ov; bidx[r] = oi;
      }
    }
  }
  if (lane == 0 || lane == 16) {
    const int rb = rowtile * 16 + (lane >> 4) * 8;
#pragma unroll
    for (int r = 0; r < 8; ++r) { wv[nhalf][rb + r] = best[r]; wi[nhalf][rb + r] = bidx[r]; }
  }
  __syncthreads();

  // Merge the two N-halves per row (lower index wins ties) + write output.
  if (tid < 64) {
    float bv = wv[0][tid];
    int bi = wi[0][tid];
    const float v1 = wv[1][tid];
    const int i1 = wi[1][tid];
    if (v1 > bv || (v1 == bv && i1 < bi)) { bv = v1; bi = i1; }
    out[(size_t)(rowbase + tid) * CBNUM + c] = bi;
  }
}

// ---------------------------------------------------------------------------
extern "C" void kernel_launch(void* const* d_in, const int* in_sizes, int n_in,
                              void* d_out, int out_size, void* d_ws, size_t ws_size,
                              hipStream_t stream) {
  (void)in_sizes; (void)n_in; (void)out_size; (void)ws_size;
  const float* x    = (const float*)d_in[0];   // (32,512,512)
  const float* proj = (const float*)d_in[1];   // (512,128)
  const float* cb   = (const float*)d_in[2];   // (8,512,128)
  int* out = (int*)d_out;                      // (32,512,1,8) int32

  // Workspace layout (44 MB total)
  float* xrec = (float*)d_ws;                               // 16384*512
  float* feat = xrec + (size_t)ROWS * IDIM;                 // 16384*128
  float* cbN  = feat + (size_t)ROWS * VQ;                   // 4096*128

  fft_rotate_kernel<<<ROWS, 256, 0, stream>>>(x, xrec);
  cbnorm_kernel<<<(CBNUM * NEMB) / 8, 256, 0, stream>>>(cb, cbN);
  proj_gemm_kernel<<<ROWS / 64, 256, 0, stream>>>(xrec, proj, feat);
  sim_argmax_kernel<<<dim3(ROWS / 64, CBNUM), 256, 0, stream>>>(feat, cbN, out);
}